// ImagePlanes_79173427134848
// MI455X (gfx1250) — compile-verified
//
#include <hip/hip_runtime.h>

#define COUNT      16
#define IMG_H      512
#define IMG_W      512
#define C_IN       6
#define C_PAD      8
#define N_PTS      262144
#define OUT_STRIDE 129          // 16*6 feats + 32 pix + 1 ts

// workspace layout (in 32-bit words from base of d_ws)
#define WS_MN_W    0            // 16 uints: float bits of per-frame min
#define WS_MX_W    16           // 16 uints: float bits of per-frame max
#define WS_P_W     64           // 48x4 floats: P = K @ pose (row = cam*3 + coord)
#define WS_FR_W    512          // frames [16][512][512][8] f32 (channel-last, padded)
#define FR_PER_FRAME ((size_t)IMG_H * (size_t)IMG_W * C_PAD)

typedef float v2f __attribute__((ext_vector_type(2)));
typedef float v8f __attribute__((ext_vector_type(8)));

// ---------------------------------------------------------------------------
// K0: compute P = K @ pose (48 rows x 4 cols) and init min/max accumulators.
// ---------------------------------------------------------------------------
__global__ __launch_bounds__(256) void k_setup(const float* __restrict__ pose,
                                               const float* __restrict__ Km,
                                               float* __restrict__ Pmat,
                                               unsigned* __restrict__ mnb,
                                               unsigned* __restrict__ mxb) {
    int t = threadIdx.x;
    if (t < 192) {                       // one P element per thread
        int row = t >> 2, col = t & 3;   // row in [0,48)
        int cam = row / 3, i = row % 3;
        const float* Kr = Km + cam * 9 + i * 3;     // K[cam][i][:]
        const float* Pc = pose + cam * 12;          // pose[cam] is [3][4]
        Pmat[row * 4 + col] = Kr[0] * Pc[0 * 4 + col]
                            + Kr[1] * Pc[1 * 4 + col]
                            + Kr[2] * Pc[2 * 4 + col];
    } else if (t < 208) {
        mnb[t - 192] = 0x7F7FFFFFu;      // +FLT_MAX bits (values are >= 0)
    } else if (t < 224) {
        mxb[t - 208] = 0u;               // 0.0f bits
    }
}

// ---------------------------------------------------------------------------
// K1: blended frames -> channel-last padded layout + per-frame min/max.
// frames[f][y][x][ch] = w1[f]*img[f][ch][y][x] + w2[f]*img[min(f+1,15)][ch][y][x]
// ---------------------------------------------------------------------------
__global__ __launch_bounds__(256) void k_blend(const float* __restrict__ img,
                                               const float* __restrict__ tc,
                                               const float* __restrict__ ts,
                                               float* __restrict__ frames,
                                               unsigned* __restrict__ mnb,
                                               unsigned* __restrict__ mxb) {
    const int f   = blockIdx.y;
    const int pix = blockIdx.x * blockDim.x + threadIdx.x;   // y*512 + x
    const float ts0 = ts[0];
    const int   f2  = (f + 1 < COUNT) ? (f + 1) : (COUNT - 1);
    const float w1  = 1.0f - fabsf(tc[f]  - ts0);
    const float w2  = 1.0f - fabsf(tc[f2] - ts0);

    const float* s1 = img + (size_t)f  * C_IN * (IMG_H * IMG_W) + pix;
    const float* s2 = img + (size_t)f2 * C_IN * (IMG_H * IMG_W) + pix;

    float v[C_IN];
    float lmin =  3.402823466e38f;
    float lmax = -3.402823466e38f;
#pragma unroll
    for (int ch = 0; ch < C_IN; ++ch) {
        float a = s1[(size_t)ch * (IMG_H * IMG_W)];
        float b = s2[(size_t)ch * (IMG_H * IMG_W)];
        float val = w1 * a + w2 * b;
        v[ch] = val;
        lmin = fminf(lmin, val);
        lmax = fmaxf(lmax, val);
    }
    float4* dst = (float4*)(frames + ((size_t)f * (IMG_H * IMG_W) + pix) * C_PAD);
    dst[0] = make_float4(v[0], v[1], v[2], v[3]);
    dst[1] = make_float4(v[4], v[5], 0.0f, 0.0f);

    // block min/max reduce, then one atomic per block (uint order == float
    // order since all values are non-negative)
    __shared__ float smin[256], smax[256];
    const int tid = threadIdx.x;
    smin[tid] = lmin; smax[tid] = lmax;
    __syncthreads();
#pragma unroll
    for (int s = 128; s > 0; s >>= 1) {
        if (tid < s) {
            smin[tid] = fminf(smin[tid], smin[tid + s]);
            smax[tid] = fmaxf(smax[tid], smax[tid + s]);
        }
        __syncthreads();
    }
    if (tid == 0) {
        atomicMin(&mnb[f], __float_as_uint(smin[0]));
        atomicMax(&mxb[f], __float_as_uint(smax[0]));
    }
}

// ---------------------------------------------------------------------------
// K2: projection via V_WMMA_F32_16X16X4_F32 + bilinear gather + fused
// normalization. Each wave owns 16 points:
//   A tiles (16x4): P rows [m*16, m*16+16)  (3 tiles cover all 48 rows)
//     lanes 0-15 hold K=0,1 ; lanes 16-31 hold K=2,3  (ISA 7.12.2)
//   B tile (4x16): 16 homogeneous points, mirrored layout
//   D (16x16): VGPR r -> (M=r, N=lane) / (M=r+8, N=lane-16)
// D is scattered through LDS so lane L then owns point (L&15) and cameras
// [0,8) or [8,16) for the sampling phase.
// ---------------------------------------------------------------------------
__global__ __launch_bounds__(256) void k_sample(const float* __restrict__ pts,
                                                const float* __restrict__ ts,
                                                const float* __restrict__ tc,
                                                const float* __restrict__ Pmat,
                                                const float* __restrict__ frames,
                                                const unsigned* __restrict__ mnb,
                                                const unsigned* __restrict__ mxb,
                                                float* __restrict__ out) {
    __shared__ float lds[8][48 * 16];    // per-wave ps tile: [row48][pt16]

    const int lane   = threadIdx.x & 31;
    const int wave   = threadIdx.x >> 5;
    const int half   = lane >> 4;        // 0 -> lanes 0-15, 1 -> lanes 16-31
    const int lane16 = lane & 15;
    const int base   = (blockIdx.x * 8 + wave) * 16;   // first point of tile

    // B tile: VGPR0 = row K=0 (half 0) / K=2 (half 1); VGPR1 = K=1 / K=3
    const float* pp = pts + (size_t)(base + lane16) * 3;
    float px0 = pp[0], px1 = pp[1], px2 = pp[2];
    v2f b;
    b.x = half ? px2 : px0;
    b.y = half ? 1.0f : px1;

#pragma unroll
    for (int m = 0; m < 3; ++m) {
        const float* pr = Pmat + (m * 16 + lane16) * 4 + half * 2;
        v2f a; a.x = pr[0]; a.y = pr[1];
        v8f c = {};
        v8f d = __builtin_amdgcn_wmma_f32_16x16x4_f32(
            /*neg_a=*/false, a, /*neg_b=*/false, b,
            /*c_mod=*/(short)0, c, /*reuse_a=*/false, /*reuse_b=*/false);
#pragma unroll
        for (int r = 0; r < 8; ++r)
            lds[wave][(m * 16 + r + half * 8) * 16 + lane16] = d[r];
    }
    __syncthreads();

    const int    pt  = base + lane16;
    float*       po  = out + (size_t)pt * OUT_STRIDE;
    const float  ts0 = ts[0];
    if (!half) po[128] = ts[pt];

    const float* wlds = lds[wave];

    for (int c8 = 0; c8 < 8; ++c8) {
        const int cam = half * 8 + c8;
        const float psx = wlds[(cam * 3 + 0) * 16 + lane16];
        const float psy = wlds[(cam * 3 + 1) * 16 + lane16];
        const float psz = wlds[(cam * 3 + 2) * 16 + lane16];
        const float inv = 1.0f / psz;

        const float gx = fminf(fmaxf(psx * inv * (1.0f / 512.0f), 0.0f), 1.0f) * 2.0f - 1.0f;
        const float gy = fminf(fmaxf(psy * inv * (1.0f / 512.0f), 0.0f), 1.0f) * 2.0f - 1.0f;
        po[96 + cam * 2 + 0] = gx;
        po[96 + cam * 2 + 1] = gy;

        // grid_sample, align_corners=False, padding_mode='zeros'
        const float x  = (gx + 1.0f) * 256.0f - 0.5f;
        const float y  = (gy + 1.0f) * 256.0f - 0.5f;
        const float xf = floorf(x), yf = floorf(y);
        const float wx1 = x - xf, wx0 = 1.0f - wx1;
        const float wy1 = y - yf, wy0 = 1.0f - wy1;
        const int   x0  = (int)xf, y0 = (int)yf;

        float a0 = 0, a1 = 0, a2 = 0, a3 = 0, a4 = 0, a5 = 0, wv = 0;
        const float* fb = frames + (size_t)cam * FR_PER_FRAME;
#pragma unroll
        for (int ty = 0; ty < 2; ++ty) {
            const int   yy = y0 + ty;
            const float wy = ty ? wy1 : wy0;
            const bool  vy = (yy >= 0) && (yy < IMG_H);
#pragma unroll
            for (int tx = 0; tx < 2; ++tx) {
                const int   xx  = x0 + tx;
                const float wgt = wy * (tx ? wx1 : wx0);
                if (vy && (xx >= 0) && (xx < IMG_W)) {
                    const float4* tp =
                        (const float4*)(fb + ((size_t)yy * IMG_W + xx) * C_PAD);
                    float4 lo = tp[0];
                    float4 hi = tp[1];
                    a0 += wgt * lo.x; a1 += wgt * lo.y; a2 += wgt * lo.z;
                    a3 += wgt * lo.w; a4 += wgt * hi.x; a5 += wgt * hi.y;
                    wv += wgt;
                }
            }
        }

        // Fused normalization: sample((v-mn)/(mx-mn)) with zero padding
        //   = (raw_sample - mn * sum(valid weights)) / (mx - mn)
        const int   cam2 = (cam + 1 < COUNT) ? (cam + 1) : (COUNT - 1);
        const float w1   = 1.0f - fabsf(tc[cam]  - ts0);
        const float w2   = 1.0f - fabsf(tc[cam2] - ts0);
        if (w1 > 0.0f || w2 > 0.0f) {
            const float mn = __uint_as_float(mnb[cam]);
            const float mx = __uint_as_float(mxb[cam]);
            const float sc = 1.0f / (mx - mn);
            const float o  = mn * wv;
            a0 = (a0 - o) * sc; a1 = (a1 - o) * sc; a2 = (a2 - o) * sc;
            a3 = (a3 - o) * sc; a4 = (a4 - o) * sc; a5 = (a5 - o) * sc;
        }
        po[cam * 6 + 0] = a0; po[cam * 6 + 1] = a1; po[cam * 6 + 2] = a2;
        po[cam * 6 + 3] = a3; po[cam * 6 + 4] = a4; po[cam * 6 + 5] = a5;
    }
}

// ---------------------------------------------------------------------------
extern "C" void kernel_launch(void* const* d_in, const int* in_sizes, int n_in,
                              void* d_out, int out_size, void* d_ws, size_t ws_size,
                              hipStream_t stream) {
    const float* pts  = (const float*)d_in[0];   // [N,3]
    const float* ts   = (const float*)d_in[1];   // [N,1]
    const float* img  = (const float*)d_in[2];   // [16,6,512,512]
    const float* pose = (const float*)d_in[3];   // [16,3,4]
    const float* Km   = (const float*)d_in[4];   // [16,3,3]
    const float* tc   = (const float*)d_in[5];   // [16]
    float*    out = (float*)d_out;
    float*    wsf = (float*)d_ws;
    unsigned* wsu = (unsigned*)d_ws;

    float*    Pmat   = wsf + WS_P_W;
    float*    frames = wsf + WS_FR_W;             // [16][512][512][8] f32
    unsigned* mnb    = wsu + WS_MN_W;
    unsigned* mxb    = wsu + WS_MX_W;

    k_setup<<<1, 256, 0, stream>>>(pose, Km, Pmat, mnb, mxb);

    dim3 gblend(IMG_H * IMG_W / 256, COUNT);
    k_blend<<<gblend, 256, 0, stream>>>(img, tc, ts, frames, mnb, mxb);

    // 256 threads = 8 waves, 16 points per wave -> 128 points per block
    k_sample<<<N_PTS / 128, 256, 0, stream>>>(pts, ts, tc, Pmat, frames,
                                              mnb, mxb, out);
}